// VectorQuantizer_56994216018336
// MI455X (gfx1250) — compile-verified
//
#include <hip/hip_runtime.h>
#include <math.h>

#define N_TOK 8192
#define VOCAB 16384
#define CDIM  256
#define NEG_INF (-1e30f)

typedef __attribute__((ext_vector_type(2))) float v2f;
typedef __attribute__((ext_vector_type(8))) float v8f;

// workspace layout (float offsets)
#define OFF_FN    0
#define OFF_CBN   (OFF_FN  + N_TOK * CDIM)   // 2097152
#define OFF_LSE   (OFF_CBN + VOCAB * CDIM)   // 6291456
#define OFF_IDX   (OFF_LSE + N_TOK)          // int[N]
#define OFF_AVG   (OFF_IDX + N_TOK)          // float[V]
#define OFF_CNT   (OFF_AVG + VOCAB)          // int[V]
#define OFF_HSUM  (OFF_CNT + VOCAB)
#define OFF_VQSUM (OFF_HSUM + 1)

// ---------------------------------------------------------------- normalize
__global__ void vq_normalize(const float* __restrict__ feat,
                             const float* __restrict__ cb,
                             float* __restrict__ fn,
                             float* __restrict__ cbn) {
  int row = blockIdx.x;
  int tid = threadIdx.x;
  const float* src; float* dst;
  if (row < N_TOK) { src = feat + (size_t)row * CDIM; dst = fn  + (size_t)row * CDIM; }
  else             { src = cb + (size_t)(row - N_TOK) * CDIM; dst = cbn + (size_t)(row - N_TOK) * CDIM; }
  float v  = src[tid];
  float sq = v * v;
  for (int off = 16; off > 0; off >>= 1) sq += __shfl_down(sq, off, 32);
  __shared__ float part[8];
  __shared__ float totS;
  int lane = tid & 31, wv = tid >> 5;
  if (lane == 0) part[wv] = sq;
  __syncthreads();
  if (tid == 0) {
    float t = 0.f;
    for (int i = 0; i < 8; ++i) t += part[i];
    totS = t;
  }
  __syncthreads();
  float r = 1.0f / fmaxf(sqrtf(totS), 1e-12f);
  dst[tid] = v * r;
}

// ------------------------------------------------------- pass 1: sim sweep
// Each workgroup: 16 token rows; 8 waves sweep V in 16-wide tiles.
// Online softmax stats + argmax kept per lane, merged via shfl_xor + LDS.
__global__ void __launch_bounds__(256) vq_pass1(const float* __restrict__ fn,
                                                const float* __restrict__ cbn,
                                                float* __restrict__ lse_out,
                                                int*   __restrict__ idx_out,
                                                float* __restrict__ hsum,
                                                float* __restrict__ vqsum) {
  __shared__ float Alds[16][CDIM + 2];
  __shared__ float Ms[8][16], Ss[8][16], Ts[8][16], Bvs[8][16];
  __shared__ int   Bis[8][16];

  const int tid  = threadIdx.x;
  const int lane = tid & 31;
  const int wv   = tid >> 5;
  const int r0   = blockIdx.x * 16;

  // cooperative A-tile load (16 rows x 256, padded stride -> conflict-free DS)
  for (int i = 0; i < 16; ++i) {
    int e = tid + i * 256;
    int rr = e >> 8, cc = e & 255;
    Alds[rr][cc] = fn[(size_t)(r0 + rr) * CDIM + cc];
  }
  __syncthreads();

  const int half = lane >> 4;     // K-pair selector per WMMA f32 layout
  const int l15  = lane & 15;     // M row (A) / N col (B)
  const int koff = half * 2;

  float m[8], s[8], t[8], bv[8];
  int bi[8];
#pragma unroll
  for (int g = 0; g < 8; ++g) { m[g] = NEG_INF; s[g] = 0.f; t[g] = 0.f; bv[g] = NEG_INF; bi[g] = 0; }

  const float* ar = &Alds[l15][koff];

  for (int tIdx = wv; tIdx < VOCAB / 16; tIdx += 8) {
    const int c0 = tIdx * 16;
    const float* brow = cbn + (size_t)(c0 + l15) * CDIM + koff;

    v8f acc0 = {0,0,0,0,0,0,0,0};
    v8f acc1 = {0,0,0,0,0,0,0,0};
#pragma unroll
    for (int kk = 0; kk < CDIM; kk += 8) {
      v2f a0 = *(const v2f*)(ar + kk);
      v2f b0 = *(const v2f*)(brow + kk);
      acc0 = __builtin_amdgcn_wmma_f32_16x16x4_f32(false, a0, false, b0, (short)0, acc0, false, false);
      v2f a1 = *(const v2f*)(ar + kk + 4);
      v2f b1 = *(const v2f*)(brow + kk + 4);
      acc1 = __builtin_amdgcn_wmma_f32_16x16x4_f32(false, a1, false, b1, (short)0, acc1, false, false);
    }
    v8f acc = acc0 + acc1;

    const int col = c0 + l15;
#pragma unroll
    for (int g = 0; g < 8; ++g) {
      float l = 200.0f * acc[g];           // logits (shifted; shift cancels)
      bool better = l > bv[g];
      bi[g] = better ? col : bi[g];
      bv[g] = better ? l : bv[g];
      float nm = fmaxf(m[g], l);
      float e0 = __expf(m[g] - nm);
      float e1 = __expf(l - nm);
      s[g] = s[g] * e0 + e1;
      t[g] = t[g] * e0 + l * e1;
      m[g] = nm;
    }
  }

  // merge across the 16 lanes of each half (butterfly)
#pragma unroll
  for (int g = 0; g < 8; ++g) {
    for (int off = 1; off < 16; off <<= 1) {
      float om  = __shfl_xor(m[g],  off, 32);
      float os  = __shfl_xor(s[g],  off, 32);
      float ot  = __shfl_xor(t[g],  off, 32);
      float obv = __shfl_xor(bv[g], off, 32);
      int   obi = __shfl_xor(bi[g], off, 32);
      float nm = fmaxf(m[g], om);
      float e0 = __expf(m[g] - nm), e1 = __expf(om - nm);
      s[g] = s[g] * e0 + os * e1;
      t[g] = t[g] * e0 + ot * e1;
      m[g] = nm;
      if (obv > bv[g] || (obv == bv[g] && obi < bi[g])) { bv[g] = obv; bi[g] = obi; }
    }
  }
  if (lane == 0 || lane == 16) {
    int rw = half * 8;
    for (int g = 0; g < 8; ++g) {
      Ms[wv][rw + g] = m[g];  Ss[wv][rw + g] = s[g];  Ts[wv][rw + g] = t[g];
      Bvs[wv][rw + g] = bv[g]; Bis[wv][rw + g] = bi[g];
    }
  }
  __syncthreads();
  if (tid < 16) {
    float fm = Ms[0][tid], fs = Ss[0][tid], ft = Ts[0][tid], fbv = Bvs[0][tid];
    int fbi = Bis[0][tid];
    for (int w2 = 1; w2 < 8; ++w2) {
      float om = Ms[w2][tid], os = Ss[w2][tid], ot = Ts[w2][tid], obv = Bvs[w2][tid];
      int obi = Bis[w2][tid];
      float nm = fmaxf(fm, om);
      float e0 = __expf(fm - nm), e1 = __expf(om - nm);
      fs = fs * e0 + os * e1;
      ft = ft * e0 + ot * e1;
      fm = nm;
      if (obv > fbv || (obv == fbv && obi < fbi)) { fbv = obv; fbi = obi; }
    }
    float lse = fm + __logf(fs);
    float H   = lse - ft / fs;              // per-sample entropy
    int row = r0 + tid;
    lse_out[row] = lse;
    idx_out[row] = fbi;
    atomicAdd(hsum, H);
    float maxsim = fbv * (1.0f / 200.0f);
    atomicAdd(vqsum, 2.0f - 2.0f * maxsim); // ||f_hat - f||^2 for unit vectors
  }
}

// --------------------------------------------- pass 2: avg_prob per column
__global__ void __launch_bounds__(256) vq_pass2(const float* __restrict__ fn,
                                                const float* __restrict__ cbn,
                                                const float* __restrict__ lse,
                                                float* __restrict__ avg) {
  __shared__ float Blds[16][CDIM + 2];
  const int tid  = threadIdx.x;
  const int lane = tid & 31;
  const int wv   = tid >> 5;
  const int c0   = blockIdx.x * 16;

  for (int i = 0; i < 16; ++i) {
    int e = tid + i * 256;
    int rr = e >> 8, cc = e & 255;
    Blds[rr][cc] = cbn[(size_t)(c0 + rr) * CDIM + cc];
  }
  __syncthreads();

  const int half = lane >> 4;
  const int l15  = lane & 15;
  const int koff = half * 2;
  const float* br = &Blds[l15][koff];

  float colsum = 0.0f;
  for (int rt = wv; rt < N_TOK / 16; rt += 8) {
    const int r0 = rt * 16;
    const float* ar = fn + (size_t)(r0 + l15) * CDIM + koff;

    v8f acc0 = {0,0,0,0,0,0,0,0};
    v8f acc1 = {0,0,0,0,0,0,0,0};
#pragma unroll
    for (int kk = 0; kk < CDIM; kk += 8) {
      v2f a0 = *(const v2f*)(ar + kk);
      v2f b0 = *(const v2f*)(br + kk);
      acc0 = __builtin_amdgcn_wmma_f32_16x16x4_f32(false, a0, false, b0, (short)0, acc0, false, false);
      v2f a1 = *(const v2f*)(ar + kk + 4);
      v2f b1 = *(const v2f*)(br + kk + 4);
      acc1 = __builtin_amdgcn_wmma_f32_16x16x4_f32(false, a1, false, b1, (short)0, acc1, false, false);
    }
    v8f acc = acc0 + acc1;
#pragma unroll
    for (int g = 0; g < 8; ++g) {
      float ls = lse[r0 + g + half * 8];
      colsum += __expf(200.0f * acc[g] - ls);
    }
  }
  atomicAdd(&avg[c0 + l15], colsum);
}

// -------------------------------------------------------------- small tail
__global__ void vq_gather(const float* __restrict__ cbn,
                          const int* __restrict__ idx,
                          float* __restrict__ out) {
  int i = blockIdx.x * 256 + threadIdx.x;
  int n = i >> 8, c = i & 255;
  out[i] = cbn[(size_t)idx[n] * CDIM + c];
}

__global__ void vq_counts(const int* __restrict__ idx, int* __restrict__ counts) {
  int i = blockIdx.x * 256 + threadIdx.x;
  atomicAdd(&counts[idx[i]], 1);
}

__global__ void vq_finalize(const float* __restrict__ avg,
                            const int* __restrict__ counts,
                            const float* __restrict__ hsum,
                            const float* __restrict__ vqsum,
                            float* __restrict__ out3) {
  int tid = threadIdx.x;
  float ce = 0.0f; int used = 0;
  for (int v = tid; v < VOCAB; v += 256) {
    float q = avg[v] * (1.0f / (float)N_TOK);
    ce += -q * __logf(q + 1e-7f);
    used += (counts[v] > 0) ? 1 : 0;
  }
  __shared__ float cf[256];
  __shared__ int   cu[256];
  cf[tid] = ce; cu[tid] = used;
  __syncthreads();
  for (int off = 128; off > 0; off >>= 1) {
    if (tid < off) { cf[tid] += cf[tid + off]; cu[tid] += cu[tid + off]; }
    __syncthreads();
  }
  if (tid == 0) {
    float vq = 1.25f * vqsum[0] / (float)(N_TOK * CDIM);
    float ps = hsum[0] / (float)N_TOK;
    out3[0] = vq;
    out3[1] = ps - cf[0];
    out3[2] = (float)cu[0] * 100.0f / (float)VOCAB;
  }
}

// ------------------------------------------------------------------ launch
extern "C" void kernel_launch(void* const* d_in, const int* in_sizes, int n_in,
                              void* d_out, int out_size, void* d_ws, size_t ws_size,
                              hipStream_t stream) {
  (void)in_sizes; (void)n_in; (void)out_size; (void)ws_size;
  const float* feat = (const float*)d_in[0];
  const float* cb   = (const float*)d_in[1];
  float* w   = (float*)d_ws;
  float* fn  = w + OFF_FN;
  float* cbn = w + OFF_CBN;
  float* lse = w + OFF_LSE;
  int*   idx = (int*)(w + OFF_IDX);
  float* avg = w + OFF_AVG;
  int*   cnt = (int*)(w + OFF_CNT);
  float* hs  = w + OFF_HSUM;
  float* vqs = w + OFF_VQSUM;
  float* out = (float*)d_out;

  // zero accumulators (avg, counts, hsum, vqsum are contiguous)
  hipMemsetAsync(avg, 0, (size_t)(2 * VOCAB + 2) * sizeof(float), stream);

  vq_normalize<<<N_TOK + VOCAB, 256, 0, stream>>>(feat, cb, fn, cbn);
  vq_pass1<<<N_TOK / 16, 256, 0, stream>>>(fn, cbn, lse, idx, hs, vqs);
  vq_gather<<<(N_TOK * CDIM) / 256, 256, 0, stream>>>(cbn, idx, out);
  vq_counts<<<N_TOK / 256, 256, 0, stream>>>(idx, cnt);
  vq_pass2<<<VOCAB / 16, 256, 0, stream>>>(fn, cbn, lse, avg);
  vq_finalize<<<1, 256, 0, stream>>>(avg, cnt, hs, vqs, out + (size_t)N_TOK * CDIM);
}